// BSplineKAN_43224550867152
// MI455X (gfx1250) — compile-verified
//
#include <hip/hip_runtime.h>
#include <stdint.h>

namespace {

constexpr int CHANNELS   = 64;
constexpr int NCP        = 8;
constexpr int ROW        = 15;                 // 3 zero | 8 cp | 3 zero | 1 pad (bank spread, gcd(15,64)=1)
constexpr int LDS_FLOATS = CHANNELS * ROW;     // 960 floats = 3840 B (240 float4s)
constexpr int BLOCK      = 256;                // 8 wave32s

typedef float v4f __attribute__((ext_vector_type(4)));

// Closed-form uniform cubic B-spline over knots linspace(-1,1,12):
//   t = (clip(x)+1)*5.5, j = floor(t), u = t-j
//   out = sum_k B_k(u) * cp[c][j-3+k];  out-of-range taps hit zero padding.
__device__ __forceinline__ float eval_bspline(const float* s_cp, int rowBase, float xv) {
    float xc = fminf(fmaxf(xv, -0.99f), 0.99f);      // -> v_med3-style clamp
    float t  = fmaf(xc, 5.5f, 5.5f);                 // (xc+1)*5.5, 5.5 exact in binary
    float jf = floorf(t);
    float u  = t - jf;
    int   j  = (int)jf;                              // 0..10
    float u2 = u * u;
    float u3 = u2 * u;
    float um = 1.0f - u;
    constexpr float k6 = 1.0f / 6.0f;
    float w0 = (um * um) * (um * k6);                // (1-u)^3/6
    float w3 = u3 * k6;                              // u^3/6
    float w1 = fmaf(0.5f, u3, (2.0f / 3.0f) - u2);   // (3u^3-6u^2+4)/6
    float w2 = fmaf(0.5f, (u2 + u) - u3, k6);        // (-3u^3+3u^2+3u+1)/6
    // padded index = (j-3+k) + 3 = j+k  ->  ds_load taps at imm offsets 0,4,8,12
    const float* row = s_cp + rowBase + j;
    return fmaf(w0, row[0], fmaf(w1, row[1], fmaf(w2, row[2], w3 * row[3])));
}

__global__ __launch_bounds__(BLOCK) void bspline_kan_kernel(
    const float* __restrict__ x,
    const float* __restrict__ cp,
    float* __restrict__ out,
    int n)
{
    __shared__ float s_cp[LDS_FLOATS];

    // Branch-simple zero of the padded table: 240 ds_store_b128.
    if (threadIdx.x < LDS_FLOATS / 4) {
        v4f z = {0.0f, 0.0f, 0.0f, 0.0f};
        *(v4f*)&s_cp[threadIdx.x * 4] = z;
    }
    __syncthreads();

    // Stage control points with CDNA5 async global->LDS DMA (tracked by ASYNCcnt).
    for (int k = threadIdx.x; k < CHANNELS * NCP; k += BLOCK) {
        int c = k >> 3;
        int i = k & 7;
        uint32_t lds_off = (uint32_t)(uintptr_t)&s_cp[c * ROW + 3 + i];  // low 32 bits of flat addr = DS offset
        uint64_t gaddr   = (uint64_t)(uintptr_t)(cp + k);
        asm volatile("global_load_async_to_lds_b32 %0, %1, off"
                     :: "v"(lds_off), "v"(gaddr) : "memory");
    }
    asm volatile("s_wait_asynccnt 0x0" ::: "memory");
    __syncthreads();

    const int n4     = n >> 2;
    const int tid    = blockIdx.x * BLOCK + threadIdx.x;
    const int stride = gridDim.x * BLOCK;

    // stride*4 is a multiple of 64, so each lane's 4 channels are loop-invariant.
    const int c0 = (tid << 2) & (CHANNELS - 1);
    const int b0 = (c0 + 0) * ROW;
    const int b1 = (c0 + 1) * ROW;
    const int b2 = (c0 + 2) * ROW;
    const int b3 = (c0 + 3) * ROW;

    const v4f* __restrict__ x4 = (const v4f*)x;
    v4f*       __restrict__ o4 = (v4f*)out;

    for (int idx = tid; idx < n4; idx += stride) {
        v4f v = __builtin_nontemporal_load(&x4[idx]);   // global_load_b128 (TH=NT, single-pass stream)
        v4f r;
        r.x = eval_bspline(s_cp, b0, v.x);
        r.y = eval_bspline(s_cp, b1, v.y);
        r.z = eval_bspline(s_cp, b2, v.z);
        r.w = eval_bspline(s_cp, b3, v.w);
        __builtin_nontemporal_store(r, &o4[idx]);       // global_store_b128 (TH=NT)
    }

    // Scalar tail (n % 4) — zero for this problem (16,777,216 floats) but kept for safety.
    int tail = n & 3;
    if (tail && tid < tail) {
        int fi = (n4 << 2) + tid;
        int c  = fi & (CHANNELS - 1);
        out[fi] = eval_bspline(s_cp, c * ROW, x[fi]);
    }
}

} // namespace

extern "C" void kernel_launch(void* const* d_in, const int* in_sizes, int n_in,
                              void* d_out, int out_size, void* d_ws, size_t ws_size,
                              hipStream_t stream) {
    const float* x  = (const float*)d_in[0];   // [262144, 64] fp32
    const float* cp = (const float*)d_in[1];   // [64, 8] fp32
    float* out      = (float*)d_out;           // [262144, 64] fp32

    int n  = in_sizes[0];
    int n4 = n >> 2;
    int blocks = (n4 + BLOCK - 1) / BLOCK;
    if (blocks > 2048) blocks = 2048;          // grid-stride; amortizes LDS staging to <2%
    if (blocks < 1) blocks = 1;

    bspline_kan_kernel<<<blocks, BLOCK, 0, stream>>>(x, cp, out, n);
}